// MoeLayer_45243185496541
// MI455X (gfx1250) — compile-verified
//
#include <hip/hip_runtime.h>
#include <hip/hip_bf16.h>

typedef __attribute__((ext_vector_type(2))) float v2f;
typedef __attribute__((ext_vector_type(8))) float v8f;

// Types for the gfx1250 async global->LDS builtin (signature from clang diagnostic:
// param0 = vector_size(16) int in AS1 (__device__), param1 = same in AS3 (__shared__)).
typedef int v4i __attribute__((vector_size(16)));
typedef __attribute__((address_space(1))) v4i gv4i;
typedef __attribute__((address_space(3))) v4i lv4i;

#define DD 1024   // model dim
#define FF 4096   // expert hidden dim
#define EE 8      // experts
#define XS_STRIDE 1028  // 16-row x tile; pad 4 -> distinct banks AND 16B-aligned rows
#define HS_STRIDE 130   // 16x128 h tile, +2 pad

#if defined(__HIP_DEVICE_COMPILE__) && __has_builtin(__builtin_amdgcn_global_load_async_to_lds_b128)
  #define ASYNC_X 1
#else
  #define ASYNC_X 0
#endif

// ---------------- kernel 0: zero output + routing counters ----------------
__global__ void moe_zero(float* __restrict__ out, int* __restrict__ cnt, int n4) {
    int i = blockIdx.x * blockDim.x + threadIdx.x;
    if (i < n4) ((float4*)out)[i] = make_float4(0.f, 0.f, 0.f, 0.f);
    if (i < EE) cnt[i] = 0;
}

// ---------------- kernel 1: gating (one wave32 per token) ----------------
__global__ void moe_gate(const float* __restrict__ x, const float* __restrict__ gw,
                         int* __restrict__ cnt, int* __restrict__ list,
                         float* __restrict__ wlist, int T) {
    int t    = blockIdx.x * 8 + (threadIdx.x >> 5);
    int lane = threadIdx.x & 31;
    if (t >= T) return;

    float acc[EE];
#pragma unroll
    for (int e = 0; e < EE; ++e) acc[e] = 0.f;

    const float* xr = x + (size_t)t * DD;
    for (int d = lane; d < DD; d += 32) {
        float xv = xr[d];
        const float* g = gw + (size_t)d * EE;   // row of 8 contiguous floats
#pragma unroll
        for (int e = 0; e < EE; ++e) acc[e] += xv * g[e];
    }
    // wave32 butterfly reduction for each of the 8 logits
#pragma unroll
    for (int e = 0; e < EE; ++e)
        for (int off = 16; off > 0; off >>= 1)
            acc[e] += __shfl_xor(acc[e], off, 32);

    if (lane == 0) {
        int e0 = 0;
        for (int e = 1; e < EE; ++e) if (acc[e] > acc[e0]) e0 = e;
        int e1 = (e0 == 0) ? 1 : 0;
        for (int e = 0; e < EE; ++e) if (e != e0 && acc[e] > acc[e1]) e1 = e;
        float v0 = acc[e0], v1 = acc[e1];
        float a0 = 1.f;                      // exp(v0 - v0)
        float a1 = __expf(v1 - v0);
        float s  = a0 + a1;
        float w0 = a0 / s, w1 = a1 / s;
        int p0 = atomicAdd(&cnt[e0], 1);
        list[e0 * T + p0]  = t;
        wlist[e0 * T + p0] = w0;
        int p1 = atomicAdd(&cnt[e1], 1);
        list[e1 * T + p1]  = t;
        wlist[e1 * T + p1] = w1;
    }
}

// ---------------- kernel 2: fused SwiGLU expert FFN via fp32 WMMA ----------------
// One block = one expert, one tile of 16 gathered tokens, 8 waves.
// Wave w owns output columns [w*128, w*128+128) as 8 persistent 16x16 accum tiles.
__global__ __launch_bounds__(256)
void moe_ffn(const float* __restrict__ x,
             const float* __restrict__ w1, const float* __restrict__ w2,
             const float* __restrict__ w3,
             const int* __restrict__ cnt, const int* __restrict__ list,
             const float* __restrict__ wlist,
             float* __restrict__ out, int T, int tilesPer) {
    __shared__ float xs[16 * XS_STRIDE];   // 16 x 1024 x-tile (padded, rows 16B aligned)
    __shared__ float hs[16 * HS_STRIDE];   // 16 x 128  h-tile (padded)
    __shared__ float wt_s[16];
    __shared__ int   tok_s[16];

    const int e  = blockIdx.x / tilesPer;
    const int mt = blockIdx.x % tilesPer;
    const int n_e = cnt[e];
    const int row0 = mt * 16;
    if (row0 >= n_e) return;

    const int tid = threadIdx.x;
    if (tid < 16) {
        int r = row0 + tid;
        bool v = r < n_e;
        tok_s[tid] = v ? list[e * T + r] : 0;
        wt_s[tid]  = v ? wlist[e * T + r] : 0.f;
    }
    __syncthreads();

    // Stage gathered 16xD activation tile into LDS.
    // CDNA5 path: async global->LDS B128 copies tracked by ASYNCcnt (no VGPR round trip).
    for (int i = tid; i < 16 * (DD / 4); i += 256) {
        int row = i >> 8;              // DD/4 = 256 float4 per row
        int c4  = i & 255;
        const float* src = x + (size_t)tok_s[row] * DD + c4 * 4;
        float*       dst = &xs[row * XS_STRIDE + c4 * 4];
#if ASYNC_X
        __builtin_amdgcn_global_load_async_to_lds_b128(
            (gv4i*)(uintptr_t)src,
            (lv4i*)(uint32_t)(uintptr_t)dst,
            0, 0);
#else
        float4 xv = *(const float4*)src;
        dst[0] = xv.x; dst[1] = xv.y; dst[2] = xv.z; dst[3] = xv.w;
#endif
    }
#if ASYNC_X
  #if __has_builtin(__builtin_amdgcn_s_wait_asynccnt)
    __builtin_amdgcn_s_wait_asynccnt(0);
  #else
    asm volatile("s_wait_asynccnt 0x0" ::: "memory");
  #endif
#endif

    const int wid  = tid >> 5;
    const int lane = tid & 31;
    const int m    = lane & 15;
    const int ksel = (lane >> 4) << 1;      // A/B frag K sub-select: 0 or 2
    const int mrowbase = (lane >> 4) << 3;  // C/D frag row base: 0 or 8

    const float* w1e = w1 + (size_t)e * DD * FF;
    const float* w3e = w3 + (size_t)e * DD * FF;
    const float* w2e = w2 + (size_t)e * FF * DD;

    v8f acc[8];
    v8f vzero = {};
#pragma unroll
    for (int i = 0; i < 8; ++i) acc[i] = vzero;

    for (int fc = 0; fc < FF; fc += 128) {
        __syncthreads();   // x-tile ready / previous chunk's GEMM2 done reading hs

        // ---- GEMM1: h = silu(x@w1) * (x@w3), this wave's 16 columns ----
        const int fcol = fc + wid * 16 + m;
        v8f h1 = vzero, h3 = vzero;
        for (int k = 0; k < DD; k += 4) {
            int kr = k + ksel;
            v2f a;
            a.x = xs[m * XS_STRIDE + kr];
            a.y = xs[m * XS_STRIDE + kr + 1];
            const float* p1 = w1e + (size_t)kr * FF + fcol;
            v2f b1; b1.x = p1[0]; b1.y = p1[FF];
            const float* p3 = w3e + (size_t)kr * FF + fcol;
            v2f b3; b3.x = p3[0]; b3.y = p3[FF];
            h1 = __builtin_amdgcn_wmma_f32_16x16x4_f32(false, a, false, b1,
                                                       (short)0, h1, false, false);
            h3 = __builtin_amdgcn_wmma_f32_16x16x4_f32(false, a, false, b3,
                                                       (short)0, h3, false, false);
        }
        // SwiGLU in registers, scatter D-frag to LDS per documented layout
#pragma unroll
        for (int j = 0; j < 8; ++j) {
            float v = h1[j];
            float g = v / (1.f + __expf(-v));
            float hv = g * h3[j];
            int row = mrowbase + j;
            hs[row * HS_STRIDE + wid * 16 + m] = hv;
        }
        __syncthreads();

        // ---- GEMM2: acc += h @ w2[fc:fc+128, :], 8 N-tiles per wave ----
#pragma unroll
        for (int nt = 0; nt < 8; ++nt) {
            const int ncol = wid * 128 + nt * 16 + m;
            for (int kk = 0; kk < 128; kk += 4) {
                int kr = kk + ksel;
                v2f a;
                a.x = hs[m * HS_STRIDE + kr];
                a.y = hs[m * HS_STRIDE + kr + 1];
                const float* p2 = w2e + (size_t)(fc + kr) * DD + ncol;
                v2f b; b.x = p2[0]; b.y = p2[DD];
                acc[nt] = __builtin_amdgcn_wmma_f32_16x16x4_f32(false, a, false, b,
                                                                (short)0, acc[nt],
                                                                false, false);
            }
        }
    }

    // ---- scale by routing weight, scatter-add into out ----
#pragma unroll
    for (int nt = 0; nt < 8; ++nt) {
#pragma unroll
        for (int j = 0; j < 8; ++j) {
            int row = mrowbase + j;
            if (row0 + row < n_e) {
                int tg = tok_s[row];
                float val = acc[nt][j] * wt_s[row];
                atomicAdd(out + (size_t)tg * DD + wid * 128 + nt * 16 + m, val);
            }
        }
    }
}

extern "C" void kernel_launch(void* const* d_in, const int* in_sizes, int n_in,
                              void* d_out, int out_size, void* d_ws, size_t ws_size,
                              hipStream_t stream) {
    const float* x      = (const float*)d_in[0];
    const float* gate_w = (const float*)d_in[1];
    const float* w1     = (const float*)d_in[2];
    const float* w2     = (const float*)d_in[3];
    const float* w3     = (const float*)d_in[4];
    float* out = (float*)d_out;

    const int T = in_sizes[0] / DD;   // 8192 tokens

    // workspace layout: [cnt: 16 ints][list: E*T ints][wlist: E*T floats]
    int*   cnt   = (int*)d_ws;
    int*   list  = cnt + 16;
    float* wlist = (float*)(list + EE * T);

    int n4 = (T * DD) / 4;
    moe_zero<<<(n4 + 255) / 256, 256, 0, stream>>>(out, cnt, n4);
    moe_gate<<<(T + 7) / 8, 256, 0, stream>>>(x, gate_w, cnt, list, wlist, T);

    int tilesPer = (T + 15) / 16;
    moe_ffn<<<EE * tilesPer, 256, 0, stream>>>(x, w1, w2, w3, cnt, list, wlist,
                                               out, T, tilesPer);
}